// WinnerTakeAll_79963701116969
// MI455X (gfx1250) — compile-verified
//
#include <hip/hip_runtime.h>
#include <stdint.h>

// Winner-take-all top-k row sparsification for MI455X (gfx1250).
// B=4096 rows, D=20000 cols, K=1000 (fp32). Memory-bound: ~656 MB total
// traffic -> ~28us floor at 23.3 TB/s. One block per row: stage the 80 KB row
// into LDS once via CDNA5 async global->LDS b128 DMA (offset-immediate fan-out,
// waited with s_wait_asynccnt), exact 4x8-bit radix select in LDS to find the
// row's 1000th-largest key, then stream the thresholded row out with b128
// stores. Ties at the threshold are kept lowest-index-first (matches
// jax.lax.top_k) via a deterministic block prefix scan.

#define ROW_D    20000
#define K_TOP    1000
#define NTHREADS 1024
#define NGROUPS  (ROW_D / 4)   // 5000 uint4 groups, exact
#define GCHUNK   5             // groups per thread in the tie pass (20 elems)

__device__ __forceinline__ uint32_t f2key(uint32_t u) {
  // monotonic map: larger float -> larger unsigned key
  return u ^ ((u & 0x80000000u) ? 0xFFFFFFFFu : 0x80000000u);
}
__device__ __forceinline__ float key2f(uint32_t k) {
  uint32_t u = k ^ ((k & 0x80000000u) ? 0x80000000u : 0xFFFFFFFFu);
  return __uint_as_float(u);
}

extern __shared__ __align__(16) uint32_t s_key[];  // ROW_D u32 = 80000 B dynamic LDS

__global__ __launch_bounds__(NTHREADS)
void wta_topk_kernel(const float* __restrict__ x, float* __restrict__ out) {
  __shared__ uint32_t s_hist[256];
  __shared__ uint32_t s_scan[NTHREADS];
  __shared__ uint32_t s_sel[2];

  const int t = threadIdx.x;
  const int row = blockIdx.x;
  const float* rowp = x + (size_t)row * ROW_D;
  float* outp = out + (size_t)row * ROW_D;
  uint4* s_key4 = (uint4*)s_key;

  // ---- stage row into LDS: CDNA5 async global->LDS b128, offset fan-out ----
  // INST_OFFSET is added to BOTH the LDS address and the global address, so
  // one (lds, vaddr) pair serves all 5 slices of this lane's data.
  {
    const uint32_t lo = (uint32_t)(uintptr_t)s_key + (uint32_t)t * 16u;
    const void* gp = (const void*)((const char*)rowp + (size_t)t * 16);
    asm volatile(
        "global_load_async_to_lds_b128 %0, %1, off\n\t"
        "global_load_async_to_lds_b128 %0, %1, off offset:16384\n\t"
        "global_load_async_to_lds_b128 %0, %1, off offset:32768\n\t"
        "global_load_async_to_lds_b128 %0, %1, off offset:49152"
        :: "v"(lo), "v"(gp) : "memory");
    if (t < NGROUPS - 4 * NTHREADS) {  // remainder slice: groups 4096..4999
      asm volatile("global_load_async_to_lds_b128 %0, %1, off offset:65536"
                   :: "v"(lo), "v"(gp) : "memory");
    }
  }
  if (t < 256) s_hist[t] = 0u;  // pass-0 histogram, zeroed while DMA flies
  asm volatile("s_wait_asynccnt 0" ::: "memory");
  __syncthreads();

  // ---- convert to order-preserving keys in place + fused pass-0 histogram ----
  for (int g = t; g < NGROUPS; g += NTHREADS) {
    uint4 k = s_key4[g];
    k.x = f2key(k.x); k.y = f2key(k.y); k.z = f2key(k.z); k.w = f2key(k.w);
    s_key4[g] = k;
    atomicAdd(&s_hist[k.x >> 24], 1u);
    atomicAdd(&s_hist[k.y >> 24], 1u);
    atomicAdd(&s_hist[k.z >> 24], 1u);
    atomicAdd(&s_hist[k.w >> 24], 1u);
  }
  __syncthreads();

  // ---- 4-pass (MSB->LSB) radix select: key of the K-th largest element ----
  uint32_t prefix = 0;
  uint32_t krem = K_TOP;
  for (int p = 0; p < 4; ++p) {
    if (p > 0) {
      if (t < 256) s_hist[t] = 0u;
      __syncthreads();
      const int shift = 24 - 8 * p;
      for (int g = t; g < NGROUPS; g += NTHREADS) {
        uint4 k = s_key4[g];
        if ((k.x >> (shift + 8)) == prefix) atomicAdd(&s_hist[(k.x >> shift) & 255u], 1u);
        if ((k.y >> (shift + 8)) == prefix) atomicAdd(&s_hist[(k.y >> shift) & 255u], 1u);
        if ((k.z >> (shift + 8)) == prefix) atomicAdd(&s_hist[(k.z >> shift) & 255u], 1u);
        if ((k.w >> (shift + 8)) == prefix) atomicAdd(&s_hist[(k.w >> shift) & 255u], 1u);
      }
      __syncthreads();
    }

    // inclusive suffix scan over 256 bins: s_scan[b] = sum_{j>=b} hist[j]
    if (t < 256) s_scan[t] = s_hist[t];
    __syncthreads();
    for (int off = 1; off < 256; off <<= 1) {
      uint32_t v = 0;
      if (t < 256 && (t + off) < 256) v = s_scan[t + off];
      __syncthreads();
      if (t < 256) s_scan[t] += v;
      __syncthreads();
    }
    // exactly one bin b satisfies suf[b+1] < krem <= suf[b]
    if (t < 256) {
      uint32_t suf  = s_scan[t];
      uint32_t suf1 = (t < 255) ? s_scan[t + 1] : 0u;
      if (suf1 < krem && krem <= suf) {
        s_sel[0] = (prefix << 8) | (uint32_t)t;
        s_sel[1] = krem - suf1;
      }
    }
    __syncthreads();
    prefix = s_sel[0];
    krem   = s_sel[1];
    __syncthreads();
  }
  const uint32_t T = prefix;  // key of the K-th largest; krem = #ties to keep

  // ---- main coalesced b128 write: all groups with no threshold ties ----
  for (int g = t; g < NGROUPS; g += NTHREADS) {
    uint4 k = s_key4[g];
    if (k.x == T || k.y == T || k.z == T || k.w == T) continue;  // tie pass owns
    float4 v;
    v.x = (k.x > T) ? key2f(k.x) : 0.0f;
    v.y = (k.y > T) ? key2f(k.y) : 0.0f;
    v.z = (k.z > T) ? key2f(k.z) : 0.0f;
    v.w = (k.w > T) ? key2f(k.w) : 0.0f;
    ((float4*)outp)[g] = v;
  }

  // ---- ties: keep first `krem` occurrences in index order (deterministic) ----
  const int gstart = t * GCHUNK;  // chunk = 20 elems = 5 groups, chunk-aligned
  uint32_t c = 0;
  if (gstart < NGROUPS) {
    for (int g = gstart; g < gstart + GCHUNK; ++g) {
      uint4 k = s_key4[g];
      c += (k.x == T) + (k.y == T) + (k.z == T) + (k.w == T);
    }
  }
  s_scan[t] = c;
  __syncthreads();
  for (int off = 1; off < NTHREADS; off <<= 1) {
    uint32_t v = 0;
    if (t >= off) v = s_scan[t - off];
    __syncthreads();
    s_scan[t] += v;
    __syncthreads();
  }
  if (gstart < NGROUPS && c > 0) {
    uint32_t base = s_scan[t] - c;  // ties in lower-index chunks
    uint32_t j = 0;
    for (int g = gstart; g < gstart + GCHUNK; ++g) {
      uint4 k = s_key4[g];
      uint32_t ks[4] = {k.x, k.y, k.z, k.w};
      bool has = (ks[0] == T) | (ks[1] == T) | (ks[2] == T) | (ks[3] == T);
      if (!has) continue;  // main pass already wrote this group
      float vs[4];
#pragma unroll
      for (int e = 0; e < 4; ++e) {
        if (ks[e] == T) {
          vs[e] = (base + j < krem) ? key2f(T) : 0.0f;
          ++j;
        } else {
          vs[e] = (ks[e] > T) ? key2f(ks[e]) : 0.0f;
        }
      }
      ((float4*)outp)[g] = make_float4(vs[0], vs[1], vs[2], vs[3]);
    }
  }
}

extern "C" void kernel_launch(void* const* d_in, const int* in_sizes, int n_in,
                              void* d_out, int out_size, void* d_ws, size_t ws_size,
                              hipStream_t stream) {
  (void)n_in; (void)d_ws; (void)ws_size; (void)out_size;
  const float* x = (const float*)d_in[0];
  float* out = (float*)d_out;
  const int B = in_sizes[0] / ROW_D;  // 4096
  dim3 grid((unsigned)B), block(NTHREADS);
  size_t dyn_lds = (size_t)ROW_D * sizeof(uint32_t);  // 80000 B per block
  wta_topk_kernel<<<grid, block, dyn_lds, stream>>>(x, out);
}